// NCutsLoss_6201932775660
// MI455X (gfx1250) — compile-verified
//
#include <hip/hip_runtime.h>
#include <stdint.h>

// Problem constants (match reference)
#define Bn 16
#define Kc 8
#define Hn 128
#define Wn 128
#define WIN 9
#define PH (Hn + WIN - 1)   // 136
#define PW (Wn + WIN - 1)   // 136
#define HW (Hn * Wn)        // 16384
#define K_CONST 8.0f

typedef __attribute__((ext_vector_type(2))) float v2f;
typedef __attribute__((ext_vector_type(8))) float v8f;

// ---------------------------------------------------------------- zero ws ---
__global__ void ncuts_zero_ws(float* ws) {
  ws[threadIdx.x] = 0.0f;   // 256 floats: gA[128] + gV[128]
}

// ------------------------------------------------------------- main kernel --
// One block per (b, h) image row: 128 threads, each owns one pixel (b,h,w=tid).
// Stage the row's weight tile (128 px * 81 taps * 4B = 41,472 B) into LDS via
// CDNA5 async global->LDS loads (fully coalesced), then each thread consumes
// its contiguous 81-float slice from LDS (stride 81 -> conflict-free banks).
__global__ __launch_bounds__(128) void ncuts_main(
    const float* __restrict__ seg, const float* __restrict__ padded,
    const float* __restrict__ weight, const float* __restrict__ sumw,
    float* __restrict__ gA, float* __restrict__ gV) {
  __shared__ float lds_w[128 * 81];
  __shared__ float red[4][16];

  const int tid = threadIdx.x;          // w coordinate
  const int b   = blockIdx.x >> 7;
  const int h   = blockIdx.x & 127;

  // ---- async stage: weight row tile -> LDS (2592 x b128 transfers) ----
  {
    const float* wbase = weight + ((size_t)b * HW + (size_t)h * Wn) * 81;
    const unsigned ldsbase = (unsigned)(uintptr_t)&lds_w[0];
#pragma unroll
    for (int i = 0; i < 21; ++i) {
      int idx = i * 128 + tid;                       // float4 index
      if (idx < (128 * 81) / 4) {
        unsigned long long ga = (unsigned long long)(uintptr_t)(wbase + idx * 4);
        unsigned la = ldsbase + (unsigned)(idx * 16);
        asm volatile("global_load_async_to_lds_b128 %0, %1, off"
                     :: "v"(la), "v"(ga) : "memory");
      }
    }
    asm volatile("s_wait_asynccnt 0x0" ::: "memory");
  }
  __syncthreads();

  // ---- windowed weighted sum: acc[k] = sum_{m,n} padded * w ----
  float acc[8];
#pragma unroll
  for (int k = 0; k < 8; ++k) acc[k] = 0.0f;

  const float* lw = &lds_w[tid * 81];
  for (int m = 0; m < 9; ++m) {
    const float* pr = padded + ((size_t)(b * Kc) * PH + (size_t)(h + m)) * PW + tid;
#pragma unroll
    for (int n = 0; n < 9; ++n) {
      float wv = lw[m * 9 + n];
#pragma unroll
      for (int k = 0; k < 8; ++k)
        acc[k] += wv * pr[(size_t)k * (PH * PW) + n];
    }
  }

  // ---- per-pixel contributions ----
  const size_t pix = (size_t)b * HW + (size_t)h * Wn + tid;
  const float sw = sumw[pix];
  float tA[8], tV[8];
#pragma unroll
  for (int k = 0; k < 8; ++k) {
    float sg = seg[(size_t)(b * Kc + k) * HW + (size_t)h * Wn + tid];
    tA[k] = acc[k] * sg;
    tV[k] = sw * sg;
  }

  // ---- wave32 shuffle reduction, then cross-wave via LDS, then atomics ----
  const int lane = tid & 31, wid = tid >> 5;
#pragma unroll
  for (int k = 0; k < 8; ++k) {
    float a = tA[k], v = tV[k];
#pragma unroll
    for (int off = 16; off > 0; off >>= 1) {
      a += __shfl_xor(a, off, 32);
      v += __shfl_xor(v, off, 32);
    }
    if (lane == 0) { red[wid][k] = a; red[wid][8 + k] = v; }
  }
  __syncthreads();
  if (tid < 16) {
    float s = red[0][tid] + red[1][tid] + red[2][tid] + red[3][tid];
    if (tid < 8) atomicAdd(&gA[b * 8 + tid], s);
    else         atomicAdd(&gV[b * 8 + (tid - 8)], s);
  }
}

// ------------------------------------------------------------ final kernel --
// out[b] = 8 - sum_k gA[b,k]/gV[b,k].  Row-sum over k done as a GEMV against
// a ones matrix with V_WMMA_F32_16X16X4_F32 (two chained WMMAs, K=4 each).
// A layout (16x4 f32): lanes 0-15 hold rows M=lane with K=0 (VGPR0), K=1
// (VGPR1); lanes 16-31 hold rows M=lane-16 with K=2,3.  D[m][j] = sum_k A[m][k].
__global__ __launch_bounds__(32) void ncuts_final(
    const float* __restrict__ gA, const float* __restrict__ gV,
    float* __restrict__ out) {
  const int lane = threadIdx.x;
  const int m  = lane & 15;
  const int kb = (lane >> 4) ? 2 : 0;   // lanes 0-15: k=0,1 ; lanes 16-31: k=2,3

  v2f A1, A2, Bones;
  A1.x = gA[m * 8 + kb + 0] / gV[m * 8 + kb + 0];
  A1.y = gA[m * 8 + kb + 1] / gV[m * 8 + kb + 1];
  A2.x = gA[m * 8 + kb + 4] / gV[m * 8 + kb + 4];   // second pass: k=4..7
  A2.y = gA[m * 8 + kb + 5] / gV[m * 8 + kb + 5];
  Bones.x = 1.0f; Bones.y = 1.0f;

  v8f d = {};
  d = __builtin_amdgcn_wmma_f32_16x16x4_f32(false, A1, false, Bones,
                                            (short)0, d, false, false);
  d = __builtin_amdgcn_wmma_f32_16x16x4_f32(false, A2, false, Bones,
                                            (short)0, d, false, false);

  // D VGPR r: lanes 0-15 = D[r][j], lanes 16-31 = D[8+r][j]
  if (lane == 0) {
#pragma unroll
    for (int r = 0; r < 8; ++r) out[r] = K_CONST - d[r];
  } else if (lane == 16) {
#pragma unroll
    for (int r = 0; r < 8; ++r) out[8 + r] = K_CONST - d[r];
  }
}

// ----------------------------------------------------------------- launch ---
extern "C" void kernel_launch(void* const* d_in, const int* in_sizes, int n_in,
                              void* d_out, int out_size, void* d_ws, size_t ws_size,
                              hipStream_t stream) {
  (void)in_sizes; (void)n_in; (void)out_size; (void)ws_size;
  const float* seg    = (const float*)d_in[0];
  const float* padded = (const float*)d_in[1];
  const float* weight = (const float*)d_in[2];
  const float* sumw   = (const float*)d_in[3];
  float* out = (float*)d_out;
  float* gA = (float*)d_ws;           // [B*K] = 128 floats
  float* gV = gA + Bn * Kc;           // [B*K] = 128 floats

  ncuts_zero_ws<<<1, 256, 0, stream>>>(gA);
  ncuts_main<<<dim3(Bn * Hn), dim3(128), 0, stream>>>(seg, padded, weight, sumw, gA, gV);
  ncuts_final<<<1, 32, 0, stream>>>(gA, gV, out);
}